// clef_multimodal_mlp_51281909514851
// MI455X (gfx1250) — compile-verified
//
#include <hip/hip_runtime.h>

typedef __attribute__((ext_vector_type(2))) float v2f;
typedef __attribute__((ext_vector_type(8))) float v8f;

#define B_SZ 128
#define L_SZ 1024
#define E_SZ 1280
#define H_SZ 128
#define DA_SZ 512
#define DB_SZ 768
#define FEAT_SZ (DA_SZ + DB_SZ)   // 1280
#define SEGS 8
#define SEG_L (L_SZ / SEGS)       // 128

// ---------------------------------------------------------------------------
// Stage 1a: ragged masked mean pool, partial sums over L-segments.
// grid (B, SEGS), block 320 threads (each thread = one float4 of the E row).
// ---------------------------------------------------------------------------
__global__ void pool_partial_kernel(const float* __restrict__ esm,
                                    const int* __restrict__ vlens,
                                    float* __restrict__ part) {
  const int b = blockIdx.x, seg = blockIdx.y;
  const int t = threadIdx.x;                       // 0..319
  const int len = vlens[b] + 2;                    // lens = valid_lens + 2
  const int l0 = seg * SEG_L;
  const int l1 = min(l0 + SEG_L, len);
  float4 acc = make_float4(0.f, 0.f, 0.f, 0.f);
  const float4* src = (const float4*)(esm + (size_t)b * L_SZ * E_SZ) + t;
  for (int l = l0; l < l1; ++l) {
    float4 v = src[(size_t)l * (E_SZ / 4)];
    acc.x += v.x; acc.y += v.y; acc.z += v.z; acc.w += v.w;
  }
  float4* dst = (float4*)(part + ((size_t)b * SEGS + seg) * E_SZ);
  dst[t] = acc;
}

// Stage 1b: reduce the 8 partials and divide by len -> x0[B,E]
__global__ void pool_reduce_kernel(const float* __restrict__ part,
                                   const int* __restrict__ vlens,
                                   float* __restrict__ x0) {
  const int b = blockIdx.x, t = threadIdx.x;       // 320 threads
  const float inv = 1.0f / (float)(vlens[b] + 2);
  float4 acc = make_float4(0.f, 0.f, 0.f, 0.f);
  for (int s = 0; s < SEGS; ++s) {
    float4 v = ((const float4*)(part + ((size_t)b * SEGS + s) * E_SZ))[t];
    acc.x += v.x; acc.y += v.y; acc.z += v.z; acc.w += v.w;
  }
  acc.x *= inv; acc.y *= inv; acc.z *= inv; acc.w *= inv;
  ((float4*)(x0 + (size_t)b * E_SZ))[t] = acc;
}

// ---------------------------------------------------------------------------
// Stage 2: fp32 WMMA GEMM  C[128,N] = act(A[128,K] @ W[K,N] + bias)
// V_WMMA_F32_16X16X4_F32, register-blocked: each wave owns one 16-col N-tile
// and FOUR 16-row M-tiles (rows 0-63 or 64-127). The B fragment is loaded
// once per k-step and reused by 4 WMMAs (4 independent accumulator chains).
// Block = 256 threads = 8 waves = {4 N-tiles} x {2 M-halves}; grid.x = N/64.
// Hot loop uses running pointers only (no per-iteration index mul).
// ---------------------------------------------------------------------------
template <bool RELU, bool BIAS>
__global__ void wmma_gemm_f32(const float* __restrict__ A,
                              const float* __restrict__ W,
                              const float* __restrict__ bias,
                              float* __restrict__ C,
                              int N, int K) {
  const int lane  = threadIdx.x & 31;
  const int wv    = threadIdx.x >> 5;          // 0..7
  const int half  = lane >> 4;                 // 0: K pair {0,1}, 1: {2,3}
  const int lr    = lane & 15;
  const int mhalf = wv & 1;                    // rows 0-63 or 64-127
  const int cn    = (blockIdx.x * 4 + (wv >> 1)) * 16;
  const int rm0   = mhalf * 64;

  v8f acc0 = {}, acc1 = {}, acc2 = {}, acc3 = {};

  // running pointers (strength-reduced)
  const float* ap = A + (size_t)(rm0 + lr) * K + 2 * half;    // A row, this lane's K pair
  const float* wp = W + (size_t)(2 * half) * N + cn + lr;     // B element, this lane's col
  const size_t arow16 = (size_t)16 * K;                       // 16 rows down
  const size_t wstep  = (size_t)4 * N;                        // next k-group

#pragma unroll 2
  for (int k = 0; k < K; k += 4) {
    v2f b;                       // B frag: lane=col, VGPR0/1 = K pair rows
    b[0] = wp[0];
    b[1] = wp[N];
    const float2 a0 = *(const float2*)(ap);
    const float2 a1 = *(const float2*)(ap + arow16);
    const float2 a2 = *(const float2*)(ap + 2 * arow16);
    const float2 a3 = *(const float2*)(ap + 3 * arow16);
    v2f va;
    va[0] = a0.x; va[1] = a0.y;
    acc0 = __builtin_amdgcn_wmma_f32_16x16x4_f32(false, va, false, b, (short)0, acc0, false, false);
    va[0] = a1.x; va[1] = a1.y;
    acc1 = __builtin_amdgcn_wmma_f32_16x16x4_f32(false, va, false, b, (short)0, acc1, false, false);
    va[0] = a2.x; va[1] = a2.y;
    acc2 = __builtin_amdgcn_wmma_f32_16x16x4_f32(false, va, false, b, (short)0, acc2, false, false);
    va[0] = a3.x; va[1] = a3.y;
    acc3 = __builtin_amdgcn_wmma_f32_16x16x4_f32(false, va, false, b, (short)0, acc3, false, false);
    ap += 4;
    wp += wstep;
  }

  const float bb = BIAS ? bias[cn + lr] : 0.0f;
  // C/D layout: VGPR i -> M = i + 8*half, N = lr
  float* crow = C + (size_t)(rm0 + 8 * half) * N + cn + lr;
  const size_t cstep = (size_t)N;
  v8f* accs[4] = {&acc0, &acc1, &acc2, &acc3};
#pragma unroll
  for (int m = 0; m < 4; ++m) {
    float* cp = crow + (size_t)m * 16 * N;
#pragma unroll
    for (int i = 0; i < 8; ++i) {
      float v = (*accs[m])[i] + bb;
      if (RELU) v = fmaxf(v, 0.0f);
      cp[(size_t)i * cstep] = v;
    }
  }
}

// ---------------------------------------------------------------------------
// Stage 3: double column-standardize of [modal_a | modal_b] -> cross[B,1280]
// One thread per column (adjacent threads -> adjacent columns -> coalesced).
// ---------------------------------------------------------------------------
__global__ void standardize2_kernel(const float* __restrict__ ma,
                                    const float* __restrict__ mb,
                                    float* __restrict__ cross) {
  const int j = blockIdx.x * blockDim.x + threadIdx.x;  // 0..1279
  const float* src; int stride, off;
  if (j < DA_SZ) { src = ma; stride = DA_SZ; off = j; }
  else           { src = mb; stride = DB_SZ; off = j - DA_SZ; }

  float sum = 0.f, sq = 0.f;
  for (int i = 0; i < B_SZ; ++i) {
    float v = src[(size_t)i * stride + off];
    sum += v; sq += v * v;
  }
  const float mu  = sum / (float)B_SZ;
  const float sd  = sqrtf(fmaxf((sq - sum * mu) / (float)(B_SZ - 1), 0.f));
  const float isd = 1.0f / sd;

  // second standardize over the standardized column (matches reference)
  float sum2 = 0.f, sq2 = 0.f;
  for (int i = 0; i < B_SZ; ++i) {
    float z = (src[(size_t)i * stride + off] - mu) * isd;
    sum2 += z; sq2 += z * z;
  }
  const float mu2  = sum2 / (float)B_SZ;
  const float sd2  = sqrtf(fmaxf((sq2 - sum2 * mu2) / (float)(B_SZ - 1), 0.f));
  const float isd2 = 1.0f / sd2;

  for (int i = 0; i < B_SZ; ++i) {
    float z = (src[(size_t)i * stride + off] - mu) * isd;
    cross[(size_t)i * FEAT_SZ + j] = (z - mu2) * isd2;
  }
}

// ---------------------------------------------------------------------------
// Stage 4: row LayerNorm over H=128 (one block of 128 threads per row)
// ---------------------------------------------------------------------------
__global__ void layernorm_kernel(const float* __restrict__ X,
                                 const float* __restrict__ g,
                                 const float* __restrict__ bta,
                                 float* __restrict__ out) {
  __shared__ float red[H_SZ];
  const int r = blockIdx.x, t = threadIdx.x;
  const float v = X[(size_t)r * H_SZ + t];
  red[t] = v; __syncthreads();
  for (int s = H_SZ / 2; s > 0; s >>= 1) {
    if (t < s) red[t] += red[t + s];
    __syncthreads();
  }
  const float mu = red[0] / (float)H_SZ;
  __syncthreads();
  const float d = v - mu;
  red[t] = d * d; __syncthreads();
  for (int s = H_SZ / 2; s > 0; s >>= 1) {
    if (t < s) red[t] += red[t + s];
    __syncthreads();
  }
  const float var = red[0] / (float)H_SZ;
  out[(size_t)r * H_SZ + t] = d * rsqrtf(var + 1e-5f) * g[t] + bta[t];
}

// ---------------------------------------------------------------------------
extern "C" void kernel_launch(void* const* d_in, const int* in_sizes, int n_in,
                              void* d_out, int out_size, void* d_ws, size_t ws_size,
                              hipStream_t stream) {
  const float* esm      = (const float*)d_in[0];
  const int*   vlens    = (const int*)d_in[1];
  const float* modal_a  = (const float*)d_in[2];
  const float* modal_b  = (const float*)d_in[3];
  const float* layer_w1 = (const float*)d_in[4];
  const float* layer_b1 = (const float*)d_in[5];
  const float* layer_w2 = (const float*)d_in[6];
  const float* layer_b2 = (const float*)d_in[7];
  const float* mlp_w1   = (const float*)d_in[8];
  const float* mlp_b1   = (const float*)d_in[9];
  const float* mlp_w2   = (const float*)d_in[10];
  const float* mlp_b2   = (const float*)d_in[11];
  const float* fe_w1    = (const float*)d_in[12];
  const float* fe_w2    = (const float*)d_in[13];
  const float* fe_b2    = (const float*)d_in[14];
  const float* lnf_g    = (const float*)d_in[15];
  const float* lnf_b    = (const float*)d_in[16];

  // workspace layout (floats)
  float* ws    = (float*)d_ws;
  float* part  = ws;                                  // 128*8*1280 = 1,310,720
  float* x0    = part + (size_t)B_SZ * SEGS * E_SZ;   // 163,840
  float* h1    = x0   + (size_t)B_SZ * E_SZ;          // 327,680
  float* h2    = h1   + (size_t)B_SZ * 2 * E_SZ;      // 327,680
  float* cross = h2   + (size_t)B_SZ * 2 * E_SZ;      // 163,840
  float* f1    = cross+ (size_t)B_SZ * FEAT_SZ;       // 131,072
  float* f2    = f1   + (size_t)B_SZ * 1024;          // 16,384

  float* out_x  = (float*)d_out;                      // [128,1280]
  float* out_px = out_x  + (size_t)B_SZ * E_SZ;       // [128,128]
  float* out_pf = out_px + (size_t)B_SZ * H_SZ;       // [128,128]

  // --- ragged mean pool (HBM-bound stage; ~671 MB streamed) ---
  pool_partial_kernel<<<dim3(B_SZ, SEGS), 320, 0, stream>>>(esm, vlens, part);
  pool_reduce_kernel<<<B_SZ, 320, 0, stream>>>(part, vlens, x0);

  // --- residual-block layer: x = relu(x0@W1+b1)@W2+b2 ---
  wmma_gemm_f32<true,  true><<<2 * E_SZ / 64, 256, 0, stream>>>(x0, layer_w1, layer_b1, h1, 2 * E_SZ, E_SZ);
  wmma_gemm_f32<false, true><<<E_SZ / 64,     256, 0, stream>>>(h1, layer_w2, layer_b2, out_x, E_SZ, 2 * E_SZ);

  // --- proj_x = relu(x@mlp_w1+b1)@mlp_w2+b2 ---
  wmma_gemm_f32<true,  true><<<2 * E_SZ / 64, 256, 0, stream>>>(out_x, mlp_w1, mlp_b1, h2, 2 * E_SZ, E_SZ);
  wmma_gemm_f32<false, true><<<H_SZ / 64,     256, 0, stream>>>(h2, mlp_w2, mlp_b2, out_px, H_SZ, 2 * E_SZ);

  // --- multimodal branch ---
  standardize2_kernel<<<FEAT_SZ / 128, 128, 0, stream>>>(modal_a, modal_b, cross);
  wmma_gemm_f32<true,  false><<<1024 / 64, 256, 0, stream>>>(cross, fe_w1, nullptr, f1, 1024, FEAT_SZ);
  wmma_gemm_f32<false, true><<<H_SZ / 64,  256, 0, stream>>>(f1, fe_w2, fe_b2, f2, H_SZ, 1024);
  layernorm_kernel<<<B_SZ, H_SZ, 0, stream>>>(f2, lnf_g, lnf_b, out_pf);
}